// GlobalAlignLoss_11184094838857
// MI455X (gfx1250) — compile-verified
//
#include <hip/hip_runtime.h>
#include <hip/hip_bf16.h>

// Chamfer/Huber loss for B=8, N=M=4096, D=3 point clouds, specialized for
// gfx1250 (CDNA5, wave32) using V_WMMA_F32_16X16X4_F32.
//
// P[b,i,j] = rx_i + ry_j - 2*dot(g_i, p_j) is produced by ONE wmma per
// 16x16 tile by folding the bias terms into the K=4 padded fragments:
//   A row i  = (-2g0, -2g1, -2g2, 1)
//   B col j  = ( p0,   p1,   p2,  ry_j)
//   C tile   =  rx_i  (broadcast per row)
// Row-mins accumulate elementwise in the D-fragment registers across the
// column loop; column-mins are obtained by re-running the same kernel with
// the two point sets swapped (distance roles are symmetric), avoiding float
// atomic-min entirely. Dimensions are compile-time constants so the
// 256-iteration column loop unrolls cleanly with no remainder path.

typedef __attribute__((ext_vector_type(2))) float v2f;
typedef __attribute__((ext_vector_type(8))) float v8f;

#define CD_B 8
#define CD_PTS 4096                 // N == M == 4096
#define CD_TILES (CD_PTS / 16)      // 256 row/column tiles
#define FLT_BIG 3.402823466e+38f

// One wave handles (batch b, row-tile itile): computes
//   rowmins[b*CD_PTS + i] = min_j ||A[b,i,:] - Bp[b,j,:]||^2
// A, Bp: [B, CD_PTS, 3].
__global__ __launch_bounds__(256)
void chamfer_rowmin_kernel(const float* __restrict__ Apts,
                           const float* __restrict__ Bpts,
                           float* __restrict__ rowmins) {
  const int lane = threadIdx.x & 31;
  const int wave = blockIdx.x * (blockDim.x >> 5) + (threadIdx.x >> 5);
  const int b     = wave / CD_TILES;
  const int itile = wave % CD_TILES;
  const int half  = lane >> 4;     // 0: lanes 0-15 (K=0,1 / rows 0-7)
  const int l16   = lane & 15;

  // ---- A-side fragment (constant across the column loop) ----
  const int m = itile * 16 + l16;
  const float* ga = Apts + ((size_t)b * CD_PTS + m) * 3;
  const float g0 = ga[0], g1 = ga[1], g2 = ga[2];
  const float rx = g0 * g0 + g1 * g1 + g2 * g2;

  v2f afrag;                                   // 16x4 f32 A: 2 VGPRs
  afrag.x = half ? (-2.0f * g2) : (-2.0f * g0);
  afrag.y = half ? 1.0f         : (-2.0f * g1);

  v8f cfrag;                                   // C[v] = rx of row (v + 8*half)
#pragma unroll
  for (int v = 0; v < 8; ++v)
    cfrag[v] = __shfl(rx, v + 8 * half, 32);

  float racc[8];
#pragma unroll
  for (int v = 0; v < 8; ++v) racc[v] = FLT_BIG;

  // ---- sweep all 256 column tiles ----
  const float* pb = Bpts + ((size_t)b * CD_PTS + l16) * 3;
  for (int jt = 0; jt < CD_TILES; ++jt, pb += 48) {  // 16 points * 3 floats
    const float p0 = pb[0], p1 = pb[1], p2 = pb[2];
    __builtin_prefetch(pb + 48, 0, 3);         // next tile, keep near (RT)
    const float ry = p0 * p0 + p1 * p1 + p2 * p2;

    v2f bfrag;                                 // 4x16 f32 B: 2 VGPRs
    bfrag.x = half ? p2 : p0;
    bfrag.y = half ? ry : p1;

    // D = A*B + C : full 16x16 distance tile in one instruction.
    v8f d = __builtin_amdgcn_wmma_f32_16x16x4_f32(
        /*neg_a=*/false, afrag, /*neg_b=*/false, bfrag,
        /*c_mod=*/(short)0, cfrag, /*reuse_a=*/false, /*reuse_b=*/false);

#pragma unroll
    for (int v = 0; v < 8; ++v) racc[v] = fminf(racc[v], d[v]);
  }

  // ---- reduce row-mins across the 16 lanes of each half ----
#pragma unroll
  for (int v = 0; v < 8; ++v) {
    float x = racc[v];
    x = fminf(x, __shfl_xor(x, 1, 32));
    x = fminf(x, __shfl_xor(x, 2, 32));
    x = fminf(x, __shfl_xor(x, 4, 32));
    x = fminf(x, __shfl_xor(x, 8, 32));
    racc[v] = x;
  }
  if (l16 == 0) {   // lane 0 -> rows 0..7, lane 16 -> rows 8..15
    float* out = rowmins + (size_t)b * CD_PTS + itile * 16 + 8 * half;
#pragma unroll
    for (int v = 0; v < 8; ++v) out[v] = racc[v];
  }
}

// Huber + global sum over both min arrays (stored contiguously), scalar out.
__global__ __launch_bounds__(256)
void huber_sum_kernel(const float* __restrict__ mins, int n,
                      const float* __restrict__ cptr,
                      float* __restrict__ out) {
  __shared__ float sdata[256];
  const float c = cptr[0];
  const float bias = 0.5f * c * c;
  float acc = 0.0f;
  for (int i = threadIdx.x; i < n; i += 256) {
    const float x = mins[i];
    acc += (x < c) ? (0.5f * x * x) : (c * x - bias);
  }
  sdata[threadIdx.x] = acc;
  __syncthreads();
  for (int s = 128; s > 0; s >>= 1) {
    if ((int)threadIdx.x < s) sdata[threadIdx.x] += sdata[threadIdx.x + s];
    __syncthreads();
  }
  if (threadIdx.x == 0) out[0] = sdata[0];
}

extern "C" void kernel_launch(void* const* d_in, const int* in_sizes, int n_in,
                              void* d_out, int out_size, void* d_ws, size_t ws_size,
                              hipStream_t stream) {
  const float* preds = (const float*)d_in[0];   // [B, M, 3]
  const float* gts   = (const float*)d_in[1];   // [B, N, 3]
  const float* c     = (const float*)d_in[2];   // scalar
  float* out = (float*)d_out;

  float* mins2 = (float*)d_ws;                    // [B*N] min over preds per gt
  float* mins1 = mins2 + (size_t)CD_B * CD_PTS;   // [B*M] min over gts per pred
  // total scratch: 2 * 8 * 4096 * 4 B = 256 KB

  // 2048 waves per pass, 8 waves/block (one 256-thread block = 8 wave32s).
  const int blocks = (CD_B * CD_TILES) / 8;

  chamfer_rowmin_kernel<<<blocks, 256, 0, stream>>>(gts, preds, mins2);
  chamfer_rowmin_kernel<<<blocks, 256, 0, stream>>>(preds, gts, mins1);
  huber_sum_kernel<<<1, 256, 0, stream>>>(mins2, CD_B * 2 * CD_PTS, c, out);
}